// GeneralizedActionFixedStackRNNG_44083544326936
// MI455X (gfx1250) — compile-verified
//
#include <hip/hip_runtime.h>
#include <stdint.h>

// ---------------------------------------------------------------------------
// GeneralizedActionFixedStack head-gather, algebraically collapsed.
//
// Per (b,s,k) item the reference's stack machine satisfies:
//   top_new >= 1 always, and write_pos == top_new - 1 in every branch
//   (both clamp cases included). The final step (t = T-1) therefore writes
//   hiddens[top_final] = nh[T-1], cells[top_final] = nc[T-1],
//   trees[top_final-1] = emb[T-1] -- exactly the positions read back.
// So: out[i] = concat(new_hiddens[T-1,i] (512), new_cells[T-1,i] (512),
//                     embs[T-1,i] (256));  actions/initials are dead inputs.
//
// => pure 3.9 MB gather, ~170 ns of HBM time at 23.3 TB/s (launch-bound).
// Zero FLOPs so WMMA is irrelevant; data movement uses the gfx1250 async
// global<->LDS DMA path (ASYNCcnt-tracked), branchless addressing.
// ---------------------------------------------------------------------------

#define T_  64
#define B_  32
#define S_  1
#define K_  8
#define H_  2
#define L_  256
#define D_  256

__global__ __launch_bounds__(320)
void rnng_head_gather_async(const float* __restrict__ embs,
                            const float* __restrict__ new_hiddens,
                            const float* __restrict__ new_cells,
                            float* __restrict__ out) {
  constexpr int HL     = H_ * L_;          // 512 floats per (H,L) slab
  constexpr int OUTW   = 2 * HL + D_;      // 1280 floats per item
  constexpr int NITEMS = B_ * S_ * K_;     // 256 items
  constexpr size_t NH_OFF  = (size_t)(T_ - 1) * NITEMS * HL;  // last timestep
  constexpr size_t EMB_OFF = (size_t)(T_ - 1) * NITEMS * D_;  // last timestep

  // 320 x 16B staging slots; each lane owns one slot and each wave only ever
  // touches its own 32 slots => per-wave s_wait_asynccnt suffices, no barrier.
  __shared__ float4 stage[320];

  const size_t i = blockIdx.x;   // item index in [0, 256)
  const int    j = threadIdx.x;  // 16-byte chunk index in [0, 320)

  // Branchless source selection (ternaries -> v_cndmask pointer selects):
  //   j in [0,128)   -> new_hiddens[T-1, i, j*4 ..]
  //   j in [128,256) -> new_cells [T-1, i, (j-128)*4 ..]
  //   j in [256,320) -> embs      [T-1, i, (j-256)*4 ..]
  const bool is_h  = j < 128;
  const bool is_hc = j < 256;
  const float* base   = is_h ? new_hiddens : (is_hc ? new_cells : embs);
  const size_t item   = is_hc ? (NH_OFF + i * HL) : (EMB_OFF + i * D_);
  const int    jl     = j - (is_h ? 0 : (is_hc ? 128 : 256));
  const float* src    = base + item + (size_t)jl * 4;
  float*       dst    = out + i * OUTW + (size_t)j * 4;

  // LDS byte address = low 32 bits of the flat address of the shared slot
  // (flat LDS addr = {shared_aperture, lds_offset32} per the ISA).
  unsigned lds_addr = (unsigned)(uintptr_t)(&stage[j]);
  unsigned long long gsrc = (unsigned long long)(uintptr_t)src;
  unsigned long long gdst = (unsigned long long)(uintptr_t)dst;

  // Async DMA: global -> LDS (16B per lane, 512B per wave per instruction).
  asm volatile("global_load_async_to_lds_b128 %0, %1, off"
               :: "v"(lds_addr), "v"(gsrc)
               : "memory");
  // Wait for this wave's async load to land in LDS (ASYNCcnt == 0).
  asm volatile("s_wait_asynccnt 0x0" ::: "memory");
  // Async DMA: LDS -> global.
  asm volatile("global_store_async_from_lds_b128 %0, %1, off"
               :: "v"(gdst), "v"(lds_addr)
               : "memory");
  // Drain before wave end (s_endpgm implies wait-idle; explicit is free).
  asm volatile("s_wait_asynccnt 0x0" ::: "memory");
}

extern "C" void kernel_launch(void* const* d_in, const int* in_sizes, int n_in,
                              void* d_out, int out_size, void* d_ws, size_t ws_size,
                              hipStream_t stream) {
  (void)in_sizes; (void)n_in; (void)out_size; (void)d_ws; (void)ws_size;

  // setup_inputs() order:
  //   0: initial_hidden (B,H,L)      -- dead (proven above)
  //   1: initial_cell   (B,H,L)      -- dead
  //   2: embs           (T,B,S,K,D)
  //   3: new_hiddens    (T,B,S,K,H,L)
  //   4: new_cells      (T,B,S,K,H,L)
  //   5: actions        (T,B,S,K)    -- dead
  const float* embs        = (const float*)d_in[2];
  const float* new_hiddens = (const float*)d_in[3];
  const float* new_cells   = (const float*)d_in[4];
  float*       out         = (float*)d_out;

  // 256 items (one block each) x 320 threads (one 16B chunk each).
  rnng_head_gather_async<<<B_ * S_ * K_, 320, 0, stream>>>(
      embs, new_hiddens, new_cells, out);
}